// SpatialConvLayer_56057913147767
// MI455X (gfx1250) — compile-verified
//
#include <hip/hip_runtime.h>
#include <hip/hip_bf16.h>

#define B_    4
#define C_IN  64
#define T_    12
#define N_    10000
#define E_    320000
#define C_OUT 64
#define ROWS  (B_ * T_ * N_)   // 480000
#define TN    (T_ * N_)        // 120000

typedef __attribute__((ext_vector_type(2))) float v2f;
typedef __attribute__((ext_vector_type(8))) float v8f;

// ---------------- utility kernels ----------------

__global__ void zero_f32(float* __restrict__ p, int n) {
    int i = blockIdx.x * blockDim.x + threadIdx.x;
    if (i < n) p[i] = 0.0f;
}

// Wc = W0 - W2 ; W1c = W1 ; W2s = 2*W2   (each 64x64, cin-major [cin*64 + cout])
__global__ void prep_weights(const float* __restrict__ W,
                             float* __restrict__ wc, float* __restrict__ w1c,
                             float* __restrict__ w2s) {
    int i = blockIdx.x * blockDim.x + threadIdx.x;
    if (i < 64 * 64) {
        float w0 = W[i];
        float w1 = W[64 * 64 + i];
        float w2 = W[2 * 64 * 64 + i];
        wc[i]  = w0 - w2;
        w1c[i] = w1;
        w2s[i] = 2.0f * w2;
    }
}

// feats0[n*64 + c] = x[b=0][c][t=0][n]  (contiguous per-node feature rows)
__global__ void gather_feats0(const float* __restrict__ x, float* __restrict__ feats0) {
    int i = blockIdx.x * blockDim.x + threadIdx.x;
    if (i < N_ * C_IN) {
        int c = i / N_;
        int n = i - c * N_;
        feats0[n * 64 + c] = x[(size_t)c * TN + n];
    }
}

__global__ void deg_kernel(const int* __restrict__ ei, float* __restrict__ deg) {
    int e = blockIdx.x * blockDim.x + threadIdx.x;
    if (e < E_) {
        int s = ei[e];                 // src row
        atomicAdd(&deg[s], 1.0f);
    }
}

__global__ void dinv_kernel(const float* __restrict__ deg, float* __restrict__ dinv) {
    int n = blockIdx.x * blockDim.x + threadIdx.x;
    if (n < N_) {
        float d = deg[n];
        dinv[n] = (d > 0.0f) ? rsqrtf(fmaxf(d, 1e-12f)) : 0.0f;
    }
}

// out[dst] += w_e * in[src] ; one wave32 per edge, 2 channels per lane
__global__ __launch_bounds__(256) void prop_kernel(const int* __restrict__ ei,
                                                   const float* __restrict__ dinv,
                                                   const float* __restrict__ fin,
                                                   float* __restrict__ fout) {
    int gid  = blockIdx.x * blockDim.x + threadIdx.x;
    int e    = gid >> 5;
    int lane = gid & 31;
    if (e < E_) {
        int s = ei[e];
        int d = ei[E_ + e];
        float w = -(dinv[s] * dinv[d]);
        const float2 v = *(const float2*)&fin[(size_t)s * 64 + 2 * lane];
        float* o = &fout[(size_t)d * 64 + 2 * lane];
        atomicAdd(o + 0, w * v.x);
        atomicAdd(o + 1, w * v.y);
    }
}

// ---------------- dense WMMA kernel ----------------
// outT(cout, n) = sum_cin W[cin][cout] * feats[n][cin]
// A-frag = W^T tile (16 cout x 4 cin) from LDS, B-frag = feats tile (4 cin x 16 n).
// Per wave: 16 rows (same b,t; consecutive n) x 64 couts -> 4 v8f accumulators.
__global__ __launch_bounds__(256) void dense_kernel(const float* __restrict__ x,
                                                    const float* __restrict__ tx1,
                                                    const float* __restrict__ pp,
                                                    const float* __restrict__ wc,
                                                    const float* __restrict__ w1c,
                                                    const float* __restrict__ w2s,
                                                    const float* __restrict__ bias,
                                                    float* __restrict__ out) {
    __shared__ float lds[3 * 64 * 64];   // 48 KB: Wc | W1 | W2s (cin-major)
    for (int i = threadIdx.x; i < 64 * 64; i += 256) {
        lds[i]               = wc[i];
        lds[64 * 64 + i]     = w1c[i];
        lds[2 * 64 * 64 + i] = w2s[i];
    }
    __syncthreads();

    const int wave = threadIdx.x >> 5;
    const int lane = threadIdx.x & 31;
    const int half = lane >> 4;     // 0: K pair {0,1} / M 0..7 ; 1: K pair {2,3} / M 8..15
    const int lid  = lane & 15;

    const int row0 = blockIdx.x * 128 + wave * 16;  // 16 consecutive rows, same (b,t)
    const int b    = row0 / TN;
    const int rem  = row0 - b * TN;
    const int t    = rem / N_;
    const int n0   = rem - t * N_;

    const float* xbase = x + (size_t)b * C_IN * TN + (size_t)t * N_ + n0;

    v8f acc[4];
#pragma unroll
    for (int ct = 0; ct < 4; ++ct) acc[ct] = (v8f){0, 0, 0, 0, 0, 0, 0, 0};

    // ---- Term 1: Tx0 @ (W0 - W2), streamed from HBM ----
    for (int kk = 0; kk < 64; kk += 4) {
        const int kA = kk + 2 * half;
        if (kk + 4 < 64)
            __builtin_prefetch(&xbase[(size_t)(kA + 4) * TN + lid], 0, 1);
        v2f bf;
        bf.x = xbase[(size_t)(kA + 0) * TN + lid];
        bf.y = xbase[(size_t)(kA + 1) * TN + lid];
#pragma unroll
        for (int ct = 0; ct < 4; ++ct) {
            v2f af;
            af.x = lds[(kA + 0) * 64 + ct * 16 + lid];
            af.y = lds[(kA + 1) * 64 + ct * 16 + lid];
            acc[ct] = __builtin_amdgcn_wmma_f32_16x16x4_f32(
                false, af, false, bf, (short)0, acc[ct], false, false);
        }
    }

    // ---- Terms 2+3 only for graph rows (wave-uniform branch; EXEC stays full) ----
    if (row0 < N_) {
        for (int kk = 0; kk < 64; kk += 4) {
            const int kA = kk + 2 * half;
            const v2f b1 = *(const v2f*)&tx1[(size_t)(n0 + lid) * 64 + kA];
            const v2f b2 = *(const v2f*)&pp[(size_t)(n0 + lid) * 64 + kA];
#pragma unroll
            for (int ct = 0; ct < 4; ++ct) {
                v2f a1, a2;
                a1.x = lds[64 * 64 + (kA + 0) * 64 + ct * 16 + lid];
                a1.y = lds[64 * 64 + (kA + 1) * 64 + ct * 16 + lid];
                a2.x = lds[2 * 64 * 64 + (kA + 0) * 64 + ct * 16 + lid];
                a2.y = lds[2 * 64 * 64 + (kA + 1) * 64 + ct * 16 + lid];
                acc[ct] = __builtin_amdgcn_wmma_f32_16x16x4_f32(
                    false, a1, false, b1, (short)0, acc[ct], false, false);
                acc[ct] = __builtin_amdgcn_wmma_f32_16x16x4_f32(
                    false, a2, false, b2, (short)0, acc[ct], false, false);
            }
        }
    }

    // ---- bias + relu + coalesced store into (B, C_OUT, T, N) ----
#pragma unroll
    for (int ct = 0; ct < 4; ++ct) {
#pragma unroll
        for (int j = 0; j < 8; ++j) {
            const int cout = ct * 16 + half * 8 + j;   // D row M = half*8 + j
            float v = acc[ct][j] + bias[cout];
            v = v > 0.0f ? v : 0.0f;
            out[(((size_t)b * C_OUT + cout) * T_ + t) * N_ + n0 + lid] = v;
        }
    }
}

// ---------------- launch ----------------

extern "C" void kernel_launch(void* const* d_in, const int* in_sizes, int n_in,
                              void* d_out, int out_size, void* d_ws, size_t ws_size,
                              hipStream_t stream) {
    (void)in_sizes; (void)n_in; (void)out_size; (void)ws_size;

    const float* x    = (const float*)d_in[0];   // (B, C_IN, T, N)
    const int*   ei   = (const int*)d_in[1];     // (2, E)
    const float* W    = (const float*)d_in[2];   // (K, C_IN, C_OUT)
    const float* bias = (const float*)d_in[3];   // (C_OUT,)
    float*       out  = (float*)d_out;           // (B, C_OUT, T, N)

    float* ws     = (float*)d_ws;
    float* deg    = ws;                          // N_
    float* dinv   = deg + N_;                    // N_
    float* feats0 = dinv + N_;                   // 64*N_
    float* tx1    = feats0 + 64 * N_;            // 64*N_  (atomic accum -> zero each call)
    float* pp     = tx1 + 64 * N_;               // 64*N_  (atomic accum -> zero each call)
    float* wc     = pp + 64 * N_;                // 4096
    float* w1c    = wc + 64 * 64;                // 4096
    float* w2s    = w1c + 64 * 64;               // 4096

    // 1) zero accumulators (must happen every launch)
    zero_f32<<<(N_ + 255) / 256, 256, 0, stream>>>(deg, N_);
    zero_f32<<<(64 * N_ + 255) / 256, 256, 0, stream>>>(tx1, 64 * N_);
    zero_f32<<<(64 * N_ + 255) / 256, 256, 0, stream>>>(pp, 64 * N_);

    // 2) folded weights
    prep_weights<<<(64 * 64 + 255) / 256, 256, 0, stream>>>(W, wc, w1c, w2s);

    // 3) contiguous node features for the graph slice
    gather_feats0<<<(N_ * C_IN + 255) / 256, 256, 0, stream>>>(x, feats0);

    // 4) degree + dinv
    deg_kernel<<<(E_ + 255) / 256, 256, 0, stream>>>(ei, deg);
    dinv_kernel<<<(N_ + 255) / 256, 256, 0, stream>>>(deg, dinv);

    // 5) Tx1 = L_hat @ Tx0 ; P = L_hat @ Tx1
    const int propThreads = E_ * 32;
    prop_kernel<<<(propThreads + 255) / 256, 256, 0, stream>>>(ei, dinv, feats0, tx1);
    prop_kernel<<<(propThreads + 255) / 256, 256, 0, stream>>>(ei, dinv, tx1, pp);

    // 6) dense WMMA GEMM + bias + relu (covers all 480000 rows, 128 per block)
    dense_kernel<<<ROWS / 128, 256, 0, stream>>>(x, tx1, pp, wc, w1c, w2s, bias, out);
}